// Yolo_loss_420906795316
// MI455X (gfx1250) — compile-verified
//
#include <hip/hip_runtime.h>

// ---------------- problem constants ----------------
#define NB 16
#define NT 60
#define NCLS 80
#define LAB_STRIDE 16
#define LAB_OFF 0
#define HM_OFF (3*NB*NT*LAB_STRIDE)      // 46080 floats
#define PART_OFF (HM_OFF + 3*NB)         // 46128 floats
#define NPART 1440                       // 1088 + 272 + 80 partial blocks

typedef __attribute__((ext_vector_type(2))) float v2f;
typedef __attribute__((ext_vector_type(8))) float v8f;

__device__ const float kAnch[18] = {12,16, 19,36, 40,28, 36,75, 76,55,
                                    72,146, 142,110, 192,243, 459,401};

__device__ __forceinline__ float clip_log(float x) {
  return fmaxf(__logf(x), -100.0f);
}

// Exact f32 wave32 sum using the matrix pipe.
// A(16x4): lane m holds A[m][0] (lanes 0-15) / A[m][2] (lanes 16-31) = v, other K slot = 0.
// B(4x16) = ones (all-ones is layout-invariant), C = 0.
// D[m][n] = v(lane m) + v(lane m+16) for every n.
// Lane L<16 has D[r][L] in d[r] (r=0..7), lane L>=16 has D[r+8][L-16].
// Sum of d[0..7] gives half-sum; fold with shfl_xor(16) -> full wave sum in all lanes.
__device__ __forceinline__ float wave_sum32(float v) {
  v2f a; a[0] = v;    a[1] = 0.0f;
  v2f b; b[0] = 1.0f; b[1] = 1.0f;
  v8f c = {};
  v8f d = __builtin_amdgcn_wmma_f32_16x16x4_f32(false, a, false, b, (short)0, c,
                                                false, false);
  float p = ((d[0]+d[1])+(d[2]+d[3])) + ((d[4]+d[5])+(d[6]+d[7]));
  p += __shfl_xor(p, 16, 32);
  return p;
}

// ---------------- stage 1: per-label precompute ----------------
__global__ void yolo_precompute(const float* __restrict__ labels,
                                float* __restrict__ lab,
                                float* __restrict__ hm) {
  int level = blockIdx.x >> 4;
  int b     = blockIdx.x & 15;
  __shared__ int sAny;
  if (threadIdx.x == 0) sAny = 0;
  __syncthreads();
  int t = threadIdx.x;
  if (t < NT) {
    const float* L = labels + (size_t)(b*NT + t)*5;
    float l0=L[0], l1=L[1], l2=L[2], l3=L[3], l4=L[4];
    bool valid = (l0+l1+l2+l3+l4) > 0.0f;
    float stride = (float)(8 << level);
    int fsize = 608 >> (3 + level);            // 76, 38, 19
    float tx = (l2+l0)/(2.0f*stride), ty = (l3+l1)/(2.0f*stride);
    float tw = (l2-l0)/stride,        th = (l3-l1)/stride;
    int best = 0; float bestv = -1e30f;
    if (valid) {
      #pragma unroll
      for (int k = 0; k < 9; k++) {
        float wb = kAnch[2*k]/stride, hb = kAnch[2*k+1]/stride;
        float iw = fminf(tw, wb), ih = fminf(th, hb);
        float ai = (iw > 0.0f && ih > 0.0f) ? iw*ih : 0.0f;
        float iou = ai / (tw*th + wb*hb - ai);
        float mw = fmaxf(tw, wb), mh = fmaxf(th, hb);
        float cc2 = mw*mw + mh*mh + 1e-16f;
        float rho2 = ((tw-wb)*(tw-wb) + (th-hb)*(th-hb))*0.25f;
        float dv = atanf(tw/th) - atanf(wb/hb);
        float vv = 0.405284735f * dv*dv;       // 4/pi^2
        float alpha = vv / (1.0f - iou + vv);
        float ci = iou - (rho2/cc2 + vv*alpha);
        if (ci > bestv) { bestv = ci; best = k; }
      }
    }
    int besta = best % 3;
    bool match = valid && ((best/3) == level);
    int ii = (int)tx; ii = ii < 0 ? 0 : (ii > fsize-1 ? fsize-1 : ii);
    int jj = (int)ty; jj = jj < 0 ? 0 : (jj > fsize-1 ? fsize-1 : jj);
    float scale = sqrtf(2.0f - tw*th/((float)fsize*(float)fsize));
    float aw = kAnch[(3*level+besta)*2]   / stride;
    float ah = kAnch[(3*level+besta)*2+1] / stride;
    float vm = valid ? 1.0f : 0.0f;
    float* o = lab + (size_t)((level*NB + b)*NT + t)*LAB_STRIDE;
    o[0] = tx*vm;  o[1] = ty*vm;
    o[2] = tw*0.5f*vm;  o[3] = th*0.5f*vm;     // half extents
    o[4] = tx - truncf(tx);  o[5] = ty - truncf(ty);
    o[6] = __logf(tw/aw + 1e-16f);
    o[7] = __logf(th/ah + 1e-16f);
    o[8] = scale;
    o[9] = match ? (float)((besta*128 + jj)*128 + ii) : -1.0f;
    o[10] = tw*th*vm;
    o[11] = l4;                                 // class id as float
    o[12] = 0.0f; o[13] = 0.0f; o[14] = 0.0f; o[15] = 0.0f;
    if (match) atomicOr(&sAny, 1);
  }
  __syncthreads();
  if (threadIdx.x == 0) hm[level*NB + b] = sAny ? 1.0f : 0.0f;
}

// ---------------- stage 2: per-level fused loss ----------------
__global__ __launch_bounds__(256)
void yolo_level(const float* __restrict__ x, const float* __restrict__ lab,
                const float* __restrict__ hm, float* __restrict__ parts,
                int level, int fsize, float stride, int partBase) {
  __shared__ float sTbx[NT], sTby[NT], sTbw2[NT], sTbh2[NT], sArea[NT], sKey[NT];
  __shared__ float sWave[8][5];
  const int b = blockIdx.y;
  const int tid = threadIdx.x;
  const float* lb = lab + (size_t)((level*NB + b)*NT)*LAB_STRIDE;
  for (int k = tid; k < NT; k += 256) {
    const float* e = lb + (size_t)k*LAB_STRIDE;
    sTbx[k]=e[0]; sTby[k]=e[1]; sTbw2[k]=e[2]; sTbh2[k]=e[3];
    sKey[k]=e[9]; sArea[k]=e[10];
  }
  __syncthreads();

  const int fs2 = fsize*fsize;
  const int cellsPerImage = 3*fs2;
  const int cell = blockIdx.x*256 + tid;
  float lxy=0.f, lwh=0.f, lobj=0.f, lcls=0.f, ll2=0.f;

  if (cell < cellsPerImage) {
    int i = cell % fsize;
    int j = (cell / fsize) % fsize;
    int a = cell / fs2;
    const float* xb = x + (size_t)(b*255 + a*85)*fs2 + (size_t)j*fsize + i;
    float o0 = xb[0];
    float o1 = xb[(size_t)fs2];
    float o2 = xb[(size_t)2*fs2];
    float o3 = xb[(size_t)3*fs2];
    float o4 = xb[(size_t)4*fs2];
    float s0   = 1.0f/(1.0f+__expf(-o0));
    float s1   = 1.0f/(1.0f+__expf(-o1));
    float sobj = 1.0f/(1.0f+__expf(-o4));
    float aw = kAnch[(3*level+a)*2]   / stride;
    float ah = kAnch[(3*level+a)*2+1] / stride;
    float px = s0 + (float)i, py = s1 + (float)j;
    float pw = __expf(o2)*aw, ph = __expf(o3)*ah;
    float pxl = px - pw*0.5f, pxr = px + pw*0.5f;
    float pyl = py - ph*0.5f, pyr = py + ph*0.5f;
    float areaP = pw*ph;
    float mykey = (float)((a*128 + j)*128 + i);
    float maxiou = 0.0f;
    int mt = -1;
    for (int t = 0; t < NT; t++) {
      float tlx = fmaxf(pxl, sTbx[t]-sTbw2[t]);
      float brx = fminf(pxr, sTbx[t]+sTbw2[t]);
      float tly = fmaxf(pyl, sTby[t]-sTbh2[t]);
      float bry = fminf(pyr, sTby[t]+sTbh2[t]);
      float dx = brx - tlx, dy = bry - tly;
      float ai = (dx > 0.0f && dy > 0.0f) ? dx*dy : 0.0f;
      float iou = ai / (areaP + sArea[t] - ai);
      maxiou = fmaxf(maxiou, iou);
      if (sKey[t] == mykey) mt = t;              // last duplicate wins
    }
    bool matched = (mt >= 0);
    float hmv = hm[level*NB + b];
    bool ignore = maxiou > 0.5f;
    float obj_mask = matched ? 1.0f : ((hmv > 0.5f) ? (ignore ? 0.0f : 1.0f) : 1.0f);
    if (matched) {
      lobj += -clip_log(sobj);
      float d = sobj - 1.0f; ll2 += d*d;
    } else if (obj_mask > 0.0f) {
      lobj += -clip_log(1.0f - sobj);
      ll2 += sobj*sobj;
    }
    if (matched) {
      const float* e = lb + (size_t)mt*LAB_STRIDE;
      float tfx=e[4], tfy=e[5], lw=e[6], lh=e[7], sc=e[8];
      int cid = (int)e[11];
      float sc2 = sc*sc;
      lxy += -(tfx*clip_log(s0) + (1.0f-tfx)*clip_log(1.0f-s0)) * sc2;
      lxy += -(tfy*clip_log(s1) + (1.0f-tfy)*clip_log(1.0f-s1)) * sc2;
      float dx0 = s0 - tfx, dy0 = s1 - tfy;
      ll2 += dx0*dx0 + dy0*dy0;
      float dw = sc*(o2 - lw), dh = sc*(o3 - lh);
      lwh += 0.5f*(dw*dw + dh*dh);
      ll2 += dw*dw + dh*dh;
      const float* xc = xb + (size_t)5*fs2;
      for (int c = 0; c < NCLS; c++) {
        float s = 1.0f/(1.0f+__expf(-xc[(size_t)c*fs2]));
        float tt = (c == cid) ? 1.0f : 0.0f;
        lcls += (c == cid) ? -clip_log(s) : -clip_log(1.0f - s);
        float dd = s - tt; ll2 += dd*dd;
      }
    }
  }

  // Block reduction: exact per-wave WMMA sums, then cross-wave via LDS.
  int wave = tid >> 5;
  int lane = tid & 31;
  float comp[5] = {lxy, lwh, lobj, lcls, ll2};
  #pragma unroll
  for (int c = 0; c < 5; c++) {
    float tot = wave_sum32(comp[c]);
    if (lane == 0) sWave[wave][c] = tot;
  }
  __syncthreads();
  if (tid < 5) {
    float s = 0.0f;
    #pragma unroll
    for (int w = 0; w < 8; w++) s += sWave[w][tid];
    parts[(size_t)(partBase + blockIdx.y*gridDim.x + blockIdx.x)*5 + tid] = s;
  }
}

// ---------------- stage 3: deterministic final reduce ----------------
__global__ void yolo_reduce(const float* __restrict__ parts, float* __restrict__ out) {
  __shared__ float sComp[5];
  int comp = threadIdx.x >> 5;     // 5 waves, one component each
  int lane = threadIdx.x & 31;
  float acc = 0.0f;
  for (int idx = lane; idx < NPART; idx += 32)   // 1440 = 45*32 -> uniform trips
    acc += parts[(size_t)idx*5 + comp];
  float tot = wave_sum32(acc);
  if (lane == 0) sComp[comp] = tot;
  __syncthreads();
  if (threadIdx.x == 0) {
    float a=sComp[0], b=sComp[1], c=sComp[2], d=sComp[3], e=sComp[4];
    out[0] = a + b + c + d;
    out[1] = a; out[2] = b; out[3] = c; out[4] = d; out[5] = e;
  }
}

extern "C" void kernel_launch(void* const* d_in, const int* in_sizes, int n_in,
                              void* d_out, int out_size, void* d_ws, size_t ws_size,
                              hipStream_t stream) {
  const float* x0     = (const float*)d_in[0];
  const float* x1     = (const float*)d_in[1];
  const float* x2     = (const float*)d_in[2];
  const float* labels = (const float*)d_in[3];
  float* wsf   = (float*)d_ws;
  float* lab   = wsf + LAB_OFF;
  float* hm    = wsf + HM_OFF;
  float* parts = wsf + PART_OFF;
  float* out   = (float*)d_out;

  yolo_precompute<<<dim3(48), dim3(64), 0, stream>>>(labels, lab, hm);
  // level 0: 3*76*76 = 17328 cells -> 68 blocks x 16 images
  yolo_level<<<dim3(68,16), dim3(256), 0, stream>>>(x0, lab, hm, parts, 0, 76,  8.0f,    0);
  // level 1: 3*38*38 = 4332 cells  -> 17 blocks
  yolo_level<<<dim3(17,16), dim3(256), 0, stream>>>(x1, lab, hm, parts, 1, 38, 16.0f, 1088);
  // level 2: 3*19*19 = 1083 cells  -> 5 blocks
  yolo_level<<<dim3(5,16),  dim3(256), 0, stream>>>(x2, lab, hm, parts, 2, 19, 32.0f, 1360);
  yolo_reduce<<<dim3(1), dim3(160), 0, stream>>>(parts, out);
}